// HardMeritBasedLoss_28965259444638
// MI455X (gfx1250) — compile-verified
//
#include <hip/hip_runtime.h>
#include <stdint.h>

// ---------------- configuration ----------------
#define THREADS  320                // 10 wave32s per row-block
#define WAVES    (THREADS / 32)
#define CHUNK    (THREADS * 4)      // 1280 floats per chunk per array (16B/lane)
#define NBUF     3                  // triple-buffered async pipeline

// T = 4.0
#define INV_T    0.25f
#define INV_T2   0.0625f
#define HALF_T2  8.0f               // 0.5 * T^2

// ---------------- CDNA5 async global -> LDS ----------------
// GV mode: vdst = LDS byte address (VGPR), vaddr = 64-bit global address (VGPR pair).
// Flat LDS pointers carry the LDS offset in their low 32 bits, so truncation yields
// the DS address the async engine needs. TH_LOAD_NT: the 524MB stream is use-once
// (2.7x the 192MB L2), so mark lines non-temporal.
__device__ __forceinline__ void async_copy_b128(const float* gptr, const float* lptr) {
    uint32_t lds_addr = (uint32_t)(uintptr_t)lptr;
    asm volatile("global_load_async_to_lds_b128 %0, %1, off th:TH_LOAD_NT"
                 :: "v"(lds_addr), "v"(gptr)
                 : "memory");
}

__global__ __launch_bounds__(THREADS)
void row_loss_kernel(const float* __restrict__ logits,
                     const float* __restrict__ targets,
                     const long long* __restrict__ t,
                     float* __restrict__ row_out,
                     int C) {
    const int row  = blockIdx.x;
    const int tid  = threadIdx.x;
    const int trow = (int)t[row];

    __shared__ __align__(16) float stage[NBUF][2][CHUNK];   // 3 x (2 x 5KB) = 30KB
    __shared__ float red_f[WAVES][6];
    __shared__ int   red_i[WAVES];
    __shared__ float s_xt;

    const float* lrow = logits  + (size_t)row * (size_t)C;
    const float* grow = targets + (size_t)row * (size_t)C;

    // online-softmax accumulators
    float mx = -3.4e38f, s1 = 0.f, s2 = 0.f;   // logits: sum exp((x-mx)/T), sum exp((x-mx)/T^2)
    float my = -3.4e38f, st = 0.f, sw = 0.f;   // targets: Z_t and sum e*(y-x)/T^2
    int   aidx = 0;                            // argmax(targets), first occurrence

    auto update = [&](float x, float y, int j) {
        if (x > mx) {
            float d = mx - x;
            s1 *= __expf(d * INV_T);
            s2 *= __expf(d * INV_T2);
            mx = x;
        }
        s1 += __expf((x - mx) * INV_T);
        s2 += __expf((x - mx) * INV_T2);
        if (y > my) {
            float e = __expf((my - y) * INV_T2);
            st *= e; sw *= e;
            my = y; aidx = j;
        }
        float e = __expf((y - my) * INV_T2);
        st += e;
        sw += e * (y - x) * INV_T2;
        if (j == trow) s_xt = x;               // exactly one lane in the block
    };

    const int nfull = C / CHUNK;

    auto issue = [&](int k) {
        const int buf = k % NBUF;
        // WAR guard: async engine must not overwrite LDS still referenced by
        // outstanding ds_loads (async LDS traffic is unordered vs DS ops).
        asm volatile("s_wait_dscnt 0" ::: "memory");
        async_copy_b128(lrow + (size_t)k * CHUNK + tid * 4, &stage[buf][0][tid * 4]);
        async_copy_b128(grow + (size_t)k * CHUNK + tid * 4, &stage[buf][1][tid * 4]);
    };

    // prologue: fill the pipeline two stages deep
    if (nfull > 0) issue(0);
    if (nfull > 1) issue(1);

    for (int k = 0; k < nfull; ++k) {
        const int cur = k % NBUF;
        if (k + 2 < nfull) {
            issue(k + 2);                                   // 6 async ops in flight
            asm volatile("s_wait_asynccnt 4" ::: "memory"); // stage k done; k+1,k+2 pending
        } else if (k + 1 < nfull) {
            asm volatile("s_wait_asynccnt 2" ::: "memory"); // drain: stage k done
        } else {
            asm volatile("s_wait_asynccnt 0" ::: "memory");
        }
        const float4 xv = *(const float4*)&stage[cur][0][tid * 4];   // ds_load_b128
        const float4 yv = *(const float4*)&stage[cur][1][tid * 4];
        const int jb = k * CHUNK + tid * 4;
        update(xv.x, yv.x, jb + 0);
        update(xv.y, yv.y, jb + 1);
        update(xv.z, yv.z, jb + 2);
        update(xv.w, yv.w, jb + 3);
    }
    // generic tail (dead for C == 32000, CHUNK == 1280)
    for (int j = nfull * CHUNK + tid; j < C; j += THREADS)
        update(lrow[j], grow[j], j);

    // ---- wave32 combine (max-rescaled sums; argmax keeps smallest index on tie) ----
    #pragma unroll
    for (int m = 16; m >= 1; m >>= 1) {
        float omx = __shfl_xor(mx, m, 32);
        float os1 = __shfl_xor(s1, m, 32);
        float os2 = __shfl_xor(s2, m, 32);
        float nm  = fmaxf(mx, omx);
        s1 = s1 * __expf((mx - nm) * INV_T)  + os1 * __expf((omx - nm) * INV_T);
        s2 = s2 * __expf((mx - nm) * INV_T2) + os2 * __expf((omx - nm) * INV_T2);
        mx = nm;

        float omy = __shfl_xor(my, m, 32);
        float ost = __shfl_xor(st, m, 32);
        float osw = __shfl_xor(sw, m, 32);
        int   oid = __shfl_xor(aidx, m, 32);
        float nmy = fmaxf(my, omy);
        float ea  = __expf((my  - nmy) * INV_T2);
        float eb  = __expf((omy - nmy) * INV_T2);
        st = st * ea + ost * eb;
        sw = sw * ea + osw * eb;
        aidx = (my > omy) ? aidx : ((omy > my) ? oid : min(aidx, oid));
        my = nmy;
    }

    const int wave = tid >> 5;
    if ((tid & 31) == 0) {
        red_f[wave][0] = mx; red_f[wave][1] = s1; red_f[wave][2] = s2;
        red_f[wave][3] = my; red_f[wave][4] = st; red_f[wave][5] = sw;
        red_i[wave]    = aidx;
    }
    __syncthreads();

    if (tid == 0) {
        mx = red_f[0][0]; s1 = red_f[0][1]; s2 = red_f[0][2];
        my = red_f[0][3]; st = red_f[0][4]; sw = red_f[0][5];
        aidx = red_i[0];
        #pragma unroll
        for (int w = 1; w < WAVES; ++w) {
            float omx = red_f[w][0], os1 = red_f[w][1], os2 = red_f[w][2];
            float omy = red_f[w][3], ost = red_f[w][4], osw = red_f[w][5];
            int   oid = red_i[w];
            float nm = fmaxf(mx, omx);
            s1 = s1 * __expf((mx - nm) * INV_T)  + os1 * __expf((omx - nm) * INV_T);
            s2 = s2 * __expf((mx - nm) * INV_T2) + os2 * __expf((omx - nm) * INV_T2);
            mx = nm;
            float nmy = fmaxf(my, omy);
            float ea = __expf((my - nmy) * INV_T2);
            float eb = __expf((omy - nmy) * INV_T2);
            st = st * ea + ost * eb;
            sw = sw * ea + osw * eb;
            aidx = (my > omy) ? aidx : ((omy > my) ? oid : min(aidx, oid));
            my = nmy;
        }
        // ce  = lse(x/T) - x_t/T
        // kld = sum p*(logp_t - logq) = sw/st + lse(x/T^2) - lse(y/T^2)
        float xt   = s_xt;
        float lse1 = mx * INV_T  + __logf(s1);
        float ce   = lse1 - xt * INV_T;
        float lse2 = mx * INV_T2 + __logf(s2);
        float lset = my * INV_T2 + __logf(st);
        float kld  = sw / st + lse2 - lset;
        row_out[row] = (aidx == trow) ? (0.5f * ce + HALF_T2 * kld) : ce;
    }
}

// Deterministic 2048 -> 1 mean (no float atomics, graph-replay stable).
__global__ __launch_bounds__(256)
void final_reduce_kernel(const float* __restrict__ row_out, float* __restrict__ out, int B) {
    __shared__ float sh[256];
    float s = 0.f;
    for (int i = threadIdx.x; i < B; i += 256) s += row_out[i];
    sh[threadIdx.x] = s;
    __syncthreads();
    #pragma unroll
    for (int m = 128; m >= 1; m >>= 1) {
        if (threadIdx.x < m) sh[threadIdx.x] += sh[threadIdx.x + m];
        __syncthreads();
    }
    if (threadIdx.x == 0) out[0] = sh[0] / (float)B;
}

extern "C" void kernel_launch(void* const* d_in, const int* in_sizes, int n_in,
                              void* d_out, int out_size, void* d_ws, size_t ws_size,
                              hipStream_t stream) {
    const float*     logits  = (const float*)d_in[0];
    const float*     targets = (const float*)d_in[1];
    const long long* t       = (const long long*)d_in[2];
    float* out = (float*)d_out;
    float* ws  = (float*)d_ws;            // B floats of scratch (8 KB)

    const int B = in_sizes[2];
    const int C = in_sizes[0] / B;

    row_loss_kernel<<<B, THREADS, 0, stream>>>(logits, targets, t, ws, C);
    final_reduce_kernel<<<1, 256, 0, stream>>>(ws, out, B);
}